// NMSLayer_81698867905088
// MI455X (gfx1250) — compile-verified
//
#include <hip/hip_runtime.h>
#include <hip/hip_bf16.h>

// Problem constants (from reference: x shape (2,1,2048,2048) float32)
#define IMG_H 2048
#define IMG_W 2048
#define NIMG  2
#define NTOT  (NIMG * IMG_H * IMG_W)          // 8388608
#define KRANK ((unsigned)((NTOT - 1) / 2))    // 4194303 -> lower median rank

// ---- CDNA5 async global->LDS path (guarded; fallback to plain loads) ----
#if defined(__AMDGCN__) && __has_builtin(__builtin_amdgcn_global_load_async_to_lds_b32) && __has_builtin(__builtin_amdgcn_s_wait_asynccnt)
#define USE_ASYNC_LDS 1
typedef __attribute__((address_space(1))) int as1i;
typedef __attribute__((address_space(3))) int as3i;
#if __has_builtin(__builtin_amdgcn_global_load_async_to_lds_b128)
#define USE_ASYNC_B128 1
typedef int v4i __attribute__((vector_size(16)));
typedef __attribute__((address_space(1))) v4i as1v4;
typedef __attribute__((address_space(3))) v4i as3v4;
#else
#define USE_ASYNC_B128 0
#endif
#else
#define USE_ASYNC_LDS 0
#define USE_ASYNC_B128 0
#endif

// Cross-kernel state in device globals (re-initialized every kernel_launch;
// no reliance on d_ws size, deterministic across graph replays).
__device__ unsigned g_bins[4096];
__device__ unsigned g_ctrl[4];  // [0]=key prefix, [1]=prefix mask, [2]=remaining rank, [3]=median float bits

// Monotone order-preserving float->uint key (IEEE total order, NaN-free input).
static __device__ __forceinline__ unsigned mkey(float f) {
  unsigned u = __float_as_uint(f);
  return (u & 0x80000000u) ? ~u : (u | 0x80000000u);
}

__global__ void init_kernel() {
  int t = threadIdx.x;
  for (int i = t; i < 4096; i += 256) g_bins[i] = 0;
  if (t == 0) {
    g_ctrl[0] = 0u;
    g_ctrl[1] = 0u;
    g_ctrl[2] = KRANK;
    g_ctrl[3] = 0u;
  }
}

// Histogram of key bits at `shift` among elements whose key matches
// the prefix selected so far. LDS-aggregated, flushed with global atomics.
__global__ __launch_bounds__(256) void hist_kernel(const float4* __restrict__ x4,
                                                   int n4, int shift, unsigned bmask) {
  __shared__ unsigned lb[4096];
  for (int i = threadIdx.x; i < 4096; i += 256) lb[i] = 0u;
  __syncthreads();

  const unsigned prefix = g_ctrl[0];
  const unsigned mask   = g_ctrl[1];
  const int stride = gridDim.x * blockDim.x;
  for (int i = blockIdx.x * blockDim.x + threadIdx.x; i < n4; i += stride) {
    float4 v = x4[i];
    unsigned k0 = mkey(v.x), k1 = mkey(v.y), k2 = mkey(v.z), k3 = mkey(v.w);
    if ((k0 & mask) == prefix) atomicAdd(&lb[(k0 >> shift) & bmask], 1u);
    if ((k1 & mask) == prefix) atomicAdd(&lb[(k1 >> shift) & bmask], 1u);
    if ((k2 & mask) == prefix) atomicAdd(&lb[(k2 >> shift) & bmask], 1u);
    if ((k3 & mask) == prefix) atomicAdd(&lb[(k3 >> shift) & bmask], 1u);
  }
  __syncthreads();
  for (int i = threadIdx.x; i < 4096; i += 256) {
    unsigned c = lb[i];
    if (c) atomicAdd(&g_bins[i], c);
  }
}

// Single-block: find the bin containing the remaining rank, extend the prefix,
// zero the bins for the next pass. Last pass reconstructs the median value.
__global__ __launch_bounds__(256) void select_kernel(int shift, int nbins, int last) {
  __shared__ unsigned lb[4096];
  for (int i = threadIdx.x; i < nbins; i += 256) lb[i] = g_bins[i];
  __syncthreads();
  if (threadIdx.x == 0) {
    unsigned r = g_ctrl[2];
    unsigned cum = 0;
    int b = nbins - 1;
    for (int i = 0; i < nbins; ++i) {
      unsigned c = lb[i];
      if (cum + c > r) { b = i; break; }
      cum += c;
    }
    g_ctrl[2] = r - cum;
    unsigned prefix = g_ctrl[0] | ((unsigned)b << shift);
    g_ctrl[0] = prefix;
    g_ctrl[1] |= ((unsigned)(nbins - 1)) << shift;
    if (last) {
      // invert monotone map: key -> float bits
      unsigned u = (prefix & 0x80000000u) ? (prefix & 0x7fffffffu) : ~prefix;
      g_ctrl[3] = u;
    }
  }
  __syncthreads();
  for (int i = threadIdx.x; i < nbins; i += 256) g_bins[i] = 0u;
}

// ---- Fused threshold + separable 7x7 max-pool + NMS compare ----
// Tile: 32x32 outputs. LDS tile origin is (by-3, bx-4): 38 rows x 40 cols.
// Column origin bx-4 keeps every 16-byte chunk aligned for async B128 DMA
// (bx is a multiple of 32 floats). Output tx reads tile cols tx+1 .. tx+7.
#define TX 32
#define TY 32
#define TCW 40  // tile cols (bx-4 .. bx+35)
#define TRH 38  // tile rows (by-3 .. by+34)

__global__ __launch_bounds__(256) void pool_kernel(const float* __restrict__ x,
                                                   float* __restrict__ out) {
  __shared__ float tile[TRH][TCW];  // raw x (or -inf halo)
  __shared__ float rmax[TRH][TX];   // horizontal 7-max of thresholded values

  const float med  = __uint_as_float(g_ctrl[3]);
  const float NINF = -__builtin_inff();

  const size_t ibase = (size_t)blockIdx.z * (IMG_H * (size_t)IMG_W);
  const float* xi = x + ibase;
  float* oi = out + ibase;

  const int bx = blockIdx.x * TX;
  const int by = blockIdx.y * TY;
  const int x0 = bx - 4, y0 = by - 3;
  const int tid = threadIdx.x;

  const bool interior = (x0 >= 0) && (y0 >= 0) && (x0 + TCW <= IMG_W) && (y0 + TRH <= IMG_H);

  if (interior) {
#if USE_ASYNC_LDS
#if USE_ASYNC_B128
    // CDNA5: 16B-per-lane async DMA global->LDS. 38 rows x 10 aligned chunks.
    for (int i = tid; i < TRH * (TCW / 4); i += 256) {
      int ty = i / (TCW / 4), c = i - ty * (TCW / 4);
      const float* gp = xi + (size_t)(y0 + ty) * IMG_W + (x0 + c * 4);
      __builtin_amdgcn_global_load_async_to_lds_b128((as1v4*)gp, (as3v4*)&tile[ty][c * 4], 0, 0);
    }
#else
    for (int i = tid; i < TRH * TCW; i += 256) {
      int ty = i / TCW, tx = i - ty * TCW;
      const float* gp = xi + (size_t)(y0 + ty) * IMG_W + (x0 + tx);
      __builtin_amdgcn_global_load_async_to_lds_b32((as1i*)gp, (as3i*)&tile[ty][tx], 0, 0);
    }
#endif
    __builtin_amdgcn_s_wait_asynccnt(0);
#else
    for (int i = tid; i < TRH * TCW; i += 256) {
      int ty = i / TCW, tx = i - ty * TCW;
      tile[ty][tx] = xi[(size_t)(y0 + ty) * IMG_W + (x0 + tx)];
    }
#endif
  } else {
    for (int i = tid; i < TRH * TCW; i += 256) {
      int ty = i / TCW, tx = i - ty * TCW;
      int gx = x0 + tx, gy = y0 + ty;
      float v = NINF;
      if (gx >= 0 && gx < IMG_W && gy >= 0 && gy < IMG_H)
        v = xi[(size_t)gy * IMG_W + gx];
      tile[ty][tx] = v;
    }
  }
  __syncthreads();

  // Horizontal 7-max with threshold applied on read. -inf halo must stay -inf
  // (reference pads with -inf BEFORE the max; threshold only applies to data).
  for (int i = tid; i < TRH * TX; i += 256) {
    int ty = i / TX, tx = i - ty * TX;
    float m = NINF;
#pragma unroll
    for (int j = 0; j < 7; ++j) {
      float v = tile[ty][tx + 1 + j];
      float t = (v < med) ? ((v == NINF) ? NINF : 0.0f) : v;
      m = fmaxf(m, t);
    }
    rmax[ty][tx] = m;
  }
  __syncthreads();

  // Vertical 7-max + compare-to-x + write.
  for (int i = tid; i < TY * TX; i += 256) {
    int ty = i / TX, tx = i - ty * TX;
    float m = rmax[ty][tx];
#pragma unroll
    for (int j = 1; j < 7; ++j) m = fmaxf(m, rmax[ty + j][tx]);
    int gx = bx + tx, gy = by + ty;
    float xv = xi[(size_t)gy * IMG_W + gx];
    float bin = (m == xv) ? 1.0f : 0.0f;  // matches reference exactly
    oi[(size_t)gy * IMG_W + gx] = bin * xv;
  }
}

extern "C" void kernel_launch(void* const* d_in, const int* in_sizes, int n_in,
                              void* d_out, int out_size, void* d_ws, size_t ws_size,
                              hipStream_t stream) {
  (void)d_ws; (void)ws_size; (void)n_in; (void)out_size;
  const float*  x  = (const float*)d_in[0];
  const float4* x4 = (const float4*)d_in[0];
  float* out = (float*)d_out;
  const int n  = in_sizes[0];   // 8388608
  const int n4 = n / 4;

  // Exact lower-median via 3-pass radix select (12+12+8 key bits).
  init_kernel<<<1, 256, 0, stream>>>();
  hist_kernel<<<1024, 256, 0, stream>>>(x4, n4, 20, 0xFFFu);
  select_kernel<<<1, 256, 0, stream>>>(20, 4096, 0);
  hist_kernel<<<1024, 256, 0, stream>>>(x4, n4, 8, 0xFFFu);
  select_kernel<<<1, 256, 0, stream>>>(8, 4096, 0);
  hist_kernel<<<1024, 256, 0, stream>>>(x4, n4, 0, 0xFFu);
  select_kernel<<<1, 256, 0, stream>>>(0, 256, 1);

  // Fused threshold + 7x7 max-pool + NMS.
  dim3 grid(IMG_W / TX, IMG_H / TY, NIMG);
  pool_kernel<<<grid, 256, 0, stream>>>(x, out);
}